// SoftmaxFreeAttention_49838800503133
// MI455X (gfx1250) — compile-verified
//
#include <hip/hip_runtime.h>
#include <math.h>

// ---------------- problem constants ----------------
#define Bn    8
#define Nn    4096
#define Dn    768
#define Hn    12
#define Mn    128
#define HDn   64
#define RATIOn 32
#define BHn   (Bn*Hn)        // 96

typedef float v2f __attribute__((ext_vector_type(2)));
typedef float v8f __attribute__((ext_vector_type(8)));

__device__ __forceinline__ v8f wmma4(v8f acc, v2f a, v2f b) {
  // V_WMMA_F32_16X16X4_F32 : D = A(16x4) * B(4x16) + C
  return __builtin_amdgcn_wmma_f32_16x16x4_f32(false, a, false, b, (short)0, acc,
                                               false, false);
}

// Accumulate a 16x16 f32 tile: acc += A(16xK) * B(Kx16).
// Element (r,k) of A at A[r*ars + k*aks]; element (k,c) of B at B[k*bks + c*bcs].
// Lane mapping per CDNA5 ISA: A row / B col = lane&15; k pair base = (lane>>4)*2.
template<int K>
__device__ __forceinline__ void gemm_tile(const float* A, int ars, int aks,
                                          const float* B, int bks, int bcs,
                                          v8f& acc, int lane) {
  const int rc = lane & 15;
  const int kb = (lane >> 4) << 1;
#pragma unroll
  for (int k = 0; k < K; k += 4) {
    v2f a, b;
    a.x = A[rc * ars + (k + kb) * aks];
    a.y = A[rc * ars + (k + kb + 1) * aks];
    b.x = B[(k + kb) * bks + rc * bcs];
    b.y = B[(k + kb + 1) * bks + rc * bcs];
    acc = wmma4(acc, a, b);
  }
}

#define Z8 {0.f,0.f,0.f,0.f,0.f,0.f,0.f,0.f}

// ======================================================================
// Kernel 1: Q/V projections.  grid (256 rowblocks, 12 heads), 256 thr.
// Qs[b,h,n,e] = (X@Wq^T + bq)*mask*HD^-0.25 ; Vb = (X@Wv^T + bv)*mask
// ======================================================================
__global__ __launch_bounds__(256) void proj_qv(
    const float* __restrict__ X, const float* __restrict__ mask,
    const float* __restrict__ Wq, const float* __restrict__ bq,
    const float* __restrict__ Wv, const float* __restrict__ bv,
    float* __restrict__ Qs, float* __restrict__ Vb, float* __restrict__ scal) {
  extern __shared__ float sm[];
  float* XT = sm;              // [128][33]
  float* BQ = XT + 128 * 33;   // [32][64]
  float* BV = BQ + 32 * 64;    // [32][64]
  const int tid = threadIdx.x, lane = tid & 31, wv = tid >> 5;
  const int h = blockIdx.y;
  const long r0 = (long)blockIdx.x * 128;  // row in [B*N)
  const int b = (int)(r0 / Nn);
  const int n0 = (int)(r0 % Nn);
  if (blockIdx.x == 0 && blockIdx.y == 0 && tid < 16) scal[tid] = 0.f;

  v8f accQ[4], accV[4];
#pragma unroll
  for (int j = 0; j < 4; j++) { v8f z = Z8; accQ[j] = z; accV[j] = z; }

  for (int kc = 0; kc < Dn; kc += 32) {
    __syncthreads();
    for (int i = tid; i < 128 * 32; i += 256) {
      int r = i >> 5, c = i & 31;
      XT[r * 33 + c] = X[(r0 + r) * Dn + kc + c];
    }
    for (int i = tid; i < 64 * 32; i += 256) {
      int j = i >> 5, k = i & 31;                    // coalesced over k
      BQ[k * 64 + j] = Wq[(long)(h * 64 + j) * Dn + kc + k];
      BV[k * 64 + j] = Wv[(long)(h * 64 + j) * Dn + kc + k];
    }
    __syncthreads();
    const float* Arow = XT + (wv * 16) * 33;
    for (int jt = 0; jt < 4; jt++) {
      gemm_tile<32>(Arow, 33, 1, BQ + jt * 16, 64, 1, accQ[jt], lane);
      gemm_tile<32>(Arow, 33, 1, BV + jt * 16, 64, 1, accV[jt], lane);
    }
  }
  const float scale = 0.35355339059327373f;  // 64^-0.25
  const int c = lane & 15, rh = (lane >> 4) * 8;
  for (int jt = 0; jt < 4; jt++) {
#pragma unroll
    for (int v = 0; v < 8; v++) {
      int r = wv * 16 + rh + v;
      int e = jt * 16 + c;
      int n = n0 + r;
      float mk = mask[(long)b * Nn + n];
      long o = (((long)(b * Hn + h)) * Nn + n) * HDn + e;
      Qs[o] = (accQ[jt][v] + bq[h * 64 + e]) * mk * scale;
      Vb[o] = (accV[jt][v] + bv[h * 64 + e]) * mk;
    }
  }
}

// ======================================================================
// Kernel 2: landmarks.  grid 96 (b*H+h).  Ql = GELU(LN(blocks x Wland))
// A[m,kk] = Qs[bh] viewed as [128][2048] (contiguous);
// B[kk,e] = Wland[e*2048 + (kk&63)*32 + (kk>>6)]
// ======================================================================
__global__ __launch_bounds__(256) void landmarks_kernel(
    const float* __restrict__ Qs, const float* __restrict__ Wland,
    const float* __restrict__ lng, const float* __restrict__ lnb,
    float* __restrict__ Ql, float* __restrict__ QlN) {
  extern __shared__ float sm[];
  float* AT = sm;              // [128][33]
  float* BT = AT + 128 * 33;   // [32][64]
  float* QB = BT + 32 * 64;    // [128][64]
  const int tid = threadIdx.x, lane = tid & 31, wv = tid >> 5;
  const int bh = blockIdx.x;
  const float* Ab = Qs + (long)bh * Nn * HDn;
  v8f acc[4];
#pragma unroll
  for (int j = 0; j < 4; j++) { v8f z = Z8; acc[j] = z; }

  for (int kc = 0; kc < RATIOn * HDn; kc += 32) {
    __syncthreads();
    for (int i = tid; i < 128 * 32; i += 256) {
      int m = i >> 5, k = i & 31;
      AT[m * 33 + k] = Ab[(long)m * 2048 + kc + k];
    }
    for (int i = tid; i < 32 * 64; i += 256) {
      int k = i & 31, e = i >> 5;
      int kk = kc + k;
      BT[k * 64 + e] = Wland[(long)e * 2048 + (kk & 63) * 32 + (kk >> 6)];
    }
    __syncthreads();
    for (int jt = 0; jt < 4; jt++)
      gemm_tile<32>(AT + wv * 16 * 33, 33, 1, BT + jt * 16, 64, 1, acc[jt], lane);
  }
  const int c = lane & 15, rh = (lane >> 4) * 8;
  for (int jt = 0; jt < 4; jt++)
#pragma unroll
    for (int v = 0; v < 8; v++)
      QB[(wv * 16 + rh + v) * 64 + jt * 16 + c] = acc[jt][v];
  __syncthreads();
  if (tid < 128) {
    int m = tid;
    float mu = 0.f, s2 = 0.f;
    for (int e = 0; e < 64; e++) mu += QB[m * 64 + e];
    mu *= (1.f / 64.f);
    for (int e = 0; e < 64; e++) { float d = QB[m * 64 + e] - mu; s2 += d * d; }
    s2 *= (1.f / 64.f);
    float rinv = rsqrtf(s2 + 1e-5f);
    float nrm = 0.f;
    for (int e = 0; e < 64; e++) {
      float y = (QB[m * 64 + e] - mu) * rinv * lng[e] + lnb[e];
      float g = 0.5f * y * (1.f + erff(y * 0.7071067811865475f));  // exact GELU
      Ql[((long)bh * Mn + m) * HDn + e] = g;
      nrm += g * g;
    }
    QlN[bh * Mn + m] = nrm;
  }
}

// ======================================================================
// Kernel 3a: k2 = exp(-max(sqdist,0)/2) per head + global max row-sum.
// ======================================================================
__global__ __launch_bounds__(256) void k2_kernel(
    const float* __restrict__ Ql, const float* __restrict__ QlN,
    float* __restrict__ K2, float* __restrict__ scal) {
  extern __shared__ float sm[];
  float* L = sm;                 // [128][65]
  float* K2b = L + 128 * 65;     // [128][128]
  float* red = K2b + 128 * 128;  // [128]
  const int tid = threadIdx.x, lane = tid & 31, wv = tid >> 5;
  const int bh = blockIdx.x;
  for (int i = tid; i < 128 * 64; i += 256) {
    int m = i >> 6, e = i & 63;
    L[m * 65 + e] = Ql[((long)bh * Mn + m) * HDn + e];
  }
  __syncthreads();
  const int c = lane & 15, rh = (lane >> 4) * 8;
  for (int ct = 0; ct < 8; ct++) {
    v8f a = Z8;
    // G = Ql @ Ql^T (K=64), B transposed
    gemm_tile<64>(L + wv * 16 * 65, 65, 1, L + ct * 16 * 65, 1, 65, a, lane);
#pragma unroll
    for (int v = 0; v < 8; v++) {
      int r = wv * 16 + rh + v, cg = ct * 16 + c;
      float d = QlN[bh * Mn + r] + QlN[bh * Mn + cg] - 2.f * a[v];
      float k2 = expf(-0.5f * fmaxf(d, 0.f));
      K2b[r * 128 + cg] = k2;
      K2[((long)bh * Mn + r) * Mn + cg] = k2;
    }
  }
  __syncthreads();
  if (tid < 128) {
    float s = 0.f;
    for (int cc = 0; cc < 128; cc++) s += K2b[tid * 128 + cc];
    red[tid] = s;
  }
  __syncthreads();
  if (tid == 0) {
    float mx = 0.f;
    for (int i = 0; i < 128; i++) mx = fmaxf(mx, red[i]);
    atomicMax((unsigned*)&scal[0], __float_as_uint(mx));  // positive floats
  }
}

__global__ void alpha0_kernel(float* scal) {
  float den = scal[0];
  scal[1] = 2.0f / (den * den);
  for (int i = 2; i < 13; i++) ((unsigned*)scal)[i] = 0u;  // pnorm slots
}

// ======================================================================
// Kernel 3c: PP = P@P in LDS; evaluate pnorm at the 11 halving candidates
// pnorm_k = max_col sum_row |I - alpha0*2^-k * PP| (global atomicMax).
// ======================================================================
__global__ __launch_bounds__(256) void pnorm_kernel(
    const float* __restrict__ K2, float* __restrict__ scal) {
  extern __shared__ float sm[];
  float* P = sm;               // [128][129]
  float* PP = P + 128 * 129;   // [128][128]
  const int tid = threadIdx.x, lane = tid & 31, wv = tid >> 5;
  const int bh = blockIdx.x;
  for (int i = tid; i < 128 * 128; i += 256) {
    int r = i >> 7, cc = i & 127;
    P[r * 129 + cc] = K2[((long)bh * Mn + r) * Mn + cc];
  }
  __syncthreads();
  const int c = lane & 15, rh = (lane >> 4) * 8;
  for (int ct = 0; ct < 8; ct++) {
    v8f a = Z8;
    gemm_tile<128>(P + wv * 16 * 129, 129, 1, P + ct * 16, 129, 1, a, lane);
#pragma unroll
    for (int v = 0; v < 8; v++)
      PP[(wv * 16 + rh + v) * 128 + ct * 16 + c] = a[v];
  }
  __syncthreads();
  float a0 = scal[1];
  if (tid < 128) {
    int cc = tid;
    float s[11];
#pragma unroll
    for (int k = 0; k < 11; k++) s[k] = 0.f;
    for (int r = 0; r < 128; r++) {
      float pp = PP[r * 128 + cc];
      float id = (r == cc) ? 1.f : 0.f;
      float al = a0;
#pragma unroll
      for (int k = 0; k < 11; k++) { s[k] += fabsf(id - al * pp); al *= 0.5f; }
    }
    for (int k = 0; k < 11; k++)
      atomicMax((unsigned*)&scal[2 + k], __float_as_uint(s[k]));
  }
}

__global__ void alpha_final_kernel(float* scal) {
  float a = scal[1];
  int j = 0;
  for (int it = 0; it < 10; it++) {
    float pn = scal[2 + j];
    if (pn > 1.01f) { a *= 0.5f; if (j < 10) j++; }
  }
  scal[13] = a;
}

// ======================================================================
// Kernel 4: Newton-Schulz, 20 iterations entirely in LDS (per b,h).
// T = P@V ; C = 2V - V@T ; blow-up guard (|C|max > 1e4 freezes V).
// ======================================================================
__global__ __launch_bounds__(256) void newton_kernel(
    const float* __restrict__ K2, const float* __restrict__ scal,
    float* __restrict__ INVm) {
  extern __shared__ float sm[];
  float* P = sm;                // [128][129]
  float* Vm = P + 128 * 129;    // [128][129]
  float* T = Vm + 128 * 129;    // [128][129]
  __shared__ unsigned bmax;
  __shared__ int okflag;
  const int tid = threadIdx.x, lane = tid & 31, wv = tid >> 5;
  const int bh = blockIdx.x;
  const float alpha = scal[13];
  for (int i = tid; i < 128 * 128; i += 256) {
    int r = i >> 7, cc = i & 127;
    float p = K2[((long)bh * Mn + r) * Mn + cc];
    P[r * 129 + cc] = p;
    Vm[r * 129 + cc] = alpha * p;
  }
  if (tid == 0) okflag = 1;
  __syncthreads();
  const int c = lane & 15, rh = (lane >> 4) * 8;
  for (int it = 0; it < 20; it++) {
    int ok0 = okflag;
    if (ok0) {
      for (int ct = 0; ct < 8; ct++) {          // T = P @ Vm
        v8f a = Z8;
        gemm_tile<128>(P + wv * 16 * 129, 129, 1, Vm + ct * 16, 129, 1, a, lane);
#pragma unroll
        for (int v = 0; v < 8; v++)
          T[(wv * 16 + rh + v) * 129 + ct * 16 + c] = a[v];
      }
    }
    __syncthreads();
    if (tid == 0) bmax = 0u;
    __syncthreads();
    v8f Cc[8];
    float lmax = 0.f;
    if (ok0) {
      for (int ct = 0; ct < 8; ct++) {          // C = 2V - V@T
        v8f a = Z8;
        gemm_tile<128>(Vm + wv * 16 * 129, 129, 1, T + ct * 16, 129, 1, a, lane);
#pragma unroll
        for (int v = 0; v < 8; v++) {
          float cv = 2.f * Vm[(wv * 16 + rh + v) * 129 + ct * 16 + c] - a[v];
          Cc[ct][v] = cv;
          lmax = fmaxf(lmax, fabsf(cv));
        }
      }
      atomicMax(&bmax, __float_as_uint(lmax));
    }
    __syncthreads();
    float bm = __uint_as_float(bmax);
    bool accept = ok0 && (bm <= 10000.f);
    __syncthreads();
    if (tid == 0 && ok0 && bm > 10000.f) okflag = 0;
    if (accept) {
      for (int ct = 0; ct < 8; ct++)
#pragma unroll
        for (int v = 0; v < 8; v++)
          Vm[(wv * 16 + rh + v) * 129 + ct * 16 + c] = Cc[ct][v];
    }
    __syncthreads();
  }
  for (int i = tid; i < 128 * 128; i += 256) {
    int r = i >> 7, cc = i & 127;
    INVm[((long)bh * Mn + r) * Mn + cc] = Vm[r * 129 + cc];
  }
}

// ======================================================================
// Kernel 5a: T = k1^T @ V with k1 computed on-the-fly (saves 400MB HBM).
// ======================================================================
__global__ __launch_bounds__(256) void ktv_kernel(
    const float* __restrict__ Qs, const float* __restrict__ Vb,
    const float* __restrict__ Ql, const float* __restrict__ QlN,
    float* __restrict__ Tg) {
  extern __shared__ float sm[];
  float* L = sm;                 // Ql  [128][65]
  float* QC = L + 128 * 65;      // Qs chunk [128][65]
  float* VC = QC + 128 * 65;     // V chunk  [128][65]
  float* K1 = VC + 128 * 65;     // [128][130]
  float* qsn = K1 + 128 * 130;   // [128]
  float* qln = qsn + 128;        // [128]
  const int tid = threadIdx.x, lane = tid & 31, wv = tid >> 5;
  const int bh = blockIdx.x;
  for (int i = tid; i < 128 * 64; i += 256) {
    int m = i >> 6, e = i & 63;
    L[m * 65 + e] = Ql[((long)bh * Mn + m) * HDn + e];
  }
  if (tid < 128) qln[tid] = QlN[bh * Mn + tid];
  v8f accT[4];
#pragma unroll
  for (int j = 0; j < 4; j++) { v8f z = Z8; accT[j] = z; }
  const int c = lane & 15, rh = (lane >> 4) * 8;
  for (int nc = 0; nc < Nn; nc += 128) {
    __syncthreads();
    for (int i = tid; i < 128 * 64; i += 256) {
      int r = i >> 6, e = i & 63;
      long o = ((long)bh * Nn + nc + r) * HDn + e;
      QC[r * 65 + e] = Qs[o];
      VC[r * 65 + e] = Vb[o];
    }
    __syncthreads();
    if (tid < 128) {
      float s = 0.f;
      for (int e = 0; e < 64; e++) { float x = QC[tid * 65 + e]; s += x * x; }
      qsn[tid] = s;
    }
    __syncthreads();
    for (int ct = 0; ct < 8; ct++) {     // G = QC @ Ql^T  ->  k1 chunk
      v8f a = Z8;
      gemm_tile<64>(QC + wv * 16 * 65, 65, 1, L + ct * 16 * 65, 1, 65, a, lane);
#pragma unroll
      for (int v = 0; v < 8; v++) {
        int r = wv * 16 + rh + v, cg = ct * 16 + c;
        float d = qsn[r] + qln[cg] - 2.f * a[v];
        K1[r * 130 + cg] = expf(-0.5f * fmaxf(d, 0.f));
      }
    }
    __syncthreads();
    for (int jt = 0; jt < 4; jt++)       // T += K1^T @ VC  (A transposed)
      gemm_tile<128>(K1 + wv * 16, 1, 130, VC + jt * 16, 65, 1, accT[jt], lane);
  }
  for (int jt = 0; jt < 4; jt++)
#pragma unroll
    for (int v = 0; v < 8; v++)
      Tg[((long)bh * Mn + wv * 16 + rh + v) * HDn + jt * 16 + c] = accT[jt][v];
}

// ======================================================================
// Kernel 5a2: U = inv @ T  (128x64, K=128) per head.
// ======================================================================
__global__ __launch_bounds__(256) void invt_kernel(
    const float* __restrict__ INVm, const float* __restrict__ Tg,
    float* __restrict__ U) {
  extern __shared__ float sm[];
  float* Iv = sm;               // [128][129]
  float* Tl = Iv + 128 * 129;   // [128][65]
  const int tid = threadIdx.x, lane = tid & 31, wv = tid >> 5;
  const int bh = blockIdx.x;
  for (int i = tid; i < 128 * 128; i += 256) {
    int r = i >> 7, cc = i & 127;
    Iv[r * 129 + cc] = INVm[((long)bh * Mn + r) * Mn + cc];
  }
  for (int i = tid; i < 128 * 64; i += 256) {
    int r = i >> 6, e = i & 63;
    Tl[r * 65 + e] = Tg[((long)bh * Mn + r) * HDn + e];
  }
  __syncthreads();
  const int c = lane & 15, rh = (lane >> 4) * 8;
  for (int jt = 0; jt < 4; jt++) {
    v8f a = Z8;
    gemm_tile<128>(Iv + wv * 16 * 129, 129, 1, Tl + jt * 16, 65, 1, a, lane);
#pragma unroll
    for (int v = 0; v < 8; v++)
      U[((long)bh * Mn + wv * 16 + rh + v) * HDn + jt * 16 + c] = a[v];
  }
}

// ======================================================================
// Kernel 5b: out = k1 @ U + depthwise conv(V), written as [b][n][h*64+e].
// ======================================================================
__global__ __launch_bounds__(256) void out_kernel(
    const float* __restrict__ Qs, const float* __restrict__ Vb,
    const float* __restrict__ Ql, const float* __restrict__ QlN,
    const float* __restrict__ U, const float* __restrict__ Wconv,
    float* __restrict__ PRE) {
  extern __shared__ float sm[];
  float* L = sm;                 // Ql [128][65]
  float* Ul = L + 128 * 65;      // U  [128][65]
  float* QC = Ul + 128 * 65;     // Qs chunk [128][65]
  float* VH = QC + 128 * 65;     // V halo [130][65]
  float* K1 = VH + 130 * 65;     // [128][130]
  float* qsn = K1 + 128 * 130;   // [128]
  float* qln = qsn + 128;        // [128]
  const int tid = threadIdx.x, lane = tid & 31, wv = tid >> 5;
  const int bh = blockIdx.x, b = bh / Hn, h = bh % Hn;
  const float w0 = Wconv[h * 3 + 0], w1 = Wconv[h * 3 + 1], w2 = Wconv[h * 3 + 2];
  for (int i = tid; i < 128 * 64; i += 256) {
    int m = i >> 6, e = i & 63;
    L[m * 65 + e] = Ql[((long)bh * Mn + m) * HDn + e];
    Ul[m * 65 + e] = U[((long)bh * Mn + m) * HDn + e];
  }
  if (tid < 128) qln[tid] = QlN[bh * Mn + tid];
  const int c = lane & 15, rh = (lane >> 4) * 8;
  for (int nc = 0; nc < Nn; nc += 128) {
    __syncthreads();
    for (int i = tid; i < 128 * 64; i += 256) {
      int r = i >> 6, e = i & 63;
      QC[r * 65 + e] = Qs[((long)bh * Nn + nc + r) * HDn + e];
    }
    for (int i = tid; i < 130 * 64; i += 256) {
      int r = i / 64, e = i % 64;
      int n = nc + r - 1;
      VH[r * 65 + e] =
          (n >= 0 && n < Nn) ? Vb[((long)bh * Nn + n) * HDn + e] : 0.f;
    }
    __syncthreads();
    if (tid < 128) {
      float s = 0.f;
      for (int e = 0; e < 64; e++) { float x = QC[tid * 65 + e]; s += x * x; }
      qsn[tid] = s;
    }
    __syncthreads();
    for (int ct = 0; ct < 8; ct++) {     // k1 chunk
      v8f a = Z8;
      gemm_tile<64>(QC + wv * 16 * 65, 65, 1, L + ct * 16 * 65, 1, 65, a, lane);
#pragma unroll
      for (int v = 0; v < 8; v++) {
        int r = wv * 16 + rh + v, cg = ct * 16 + c;
        float d = qsn[r] + qln[cg] - 2.f * a[v];
        K1[r * 130 + cg] = expf(-0.5f * fmaxf(d, 0.f));
      }
    }
    __syncthreads();
    for (int jt = 0; jt < 4; jt++) {     // O = K1 @ U  (+ conv)
      v8f a = Z8;
      gemm_tile<128>(K1 + wv * 16 * 130, 130, 1, Ul + jt * 16, 65, 1, a, lane);
#pragma unroll
      for (int v = 0; v < 8; v++) {
        int r = wv * 16 + rh + v, e = jt * 16 + c;
        float cv = w0 * VH[r * 65 + e] + w1 * VH[(r + 1) * 65 + e] +
                   w2 * VH[(r + 2) * 65 + e];
        int n = nc + r;
        PRE[((long)b * Nn + n) * Dn + h * 64 + e] = a[v] + cv;
      }
    }
  }
}

// ======================================================================
// Kernel 6: final projection  Y = PRE @ Wo^T + bo.
// ======================================================================
__global__ __launch_bounds__(256) void outproj_kernel(
    const float* __restrict__ PRE, const float* __restrict__ Wo,
    const float* __restrict__ bo, float* __restrict__ Y) {
  extern __shared__ float sm[];
  float* AT = sm;              // [128][33]
  float* BT = AT + 128 * 33;   // [32][64]
  const int tid = threadIdx.x, lane = tid & 31, wv = tid >> 5;
  const long r0 = (long)blockIdx.x * 128;  // rows over B*N
  const int jb = blockIdx.y;               // 64-wide column block
  v8f acc[4];
#pragma unroll
  for (int j = 0; j < 4; j++) { v8f z = Z8; acc[j] = z; }
  for (int kc = 0; kc < Dn; kc += 32) {
    __syncthreads();
    for (int i = tid; i < 128 * 32; i += 256) {
      int r = i >> 5, cc = i & 31;
      AT[r * 33 + cc] = PRE[(r0 + r) * Dn + kc + cc];
    }
    for (int i = tid; i < 64 * 32; i += 256) {
      int j = i >> 5, k = i & 31;
      BT[k * 64 + j] = Wo[(long)(jb * 64 + j) * Dn + kc + k];
    }
    __syncthreads();
    for (int jt = 0; jt < 4; jt++)
      gemm_tile<32>(AT + wv * 16 * 33, 33, 1, BT + jt * 16, 64, 1, acc[jt], lane);
  }
  const int c = lane & 15, rh = (lane >> 4) * 8;
  for (int jt = 0; jt < 4; jt++)
#pragma unroll
    for (int v = 0; v < 8; v++) {
      int e = jb * 64 + jt * 16 + c;
      Y[(r0 + wv * 16 + rh + v) * Dn + e] = acc[jt][v] + bo[e];
    }
}

// ======================================================================
extern "C" void kernel_launch(void* const* d_in, const int* in_sizes, int n_in,
                              void* d_out, int out_size, void* d_ws,
                              size_t ws_size, hipStream_t stream) {
  (void)in_sizes; (void)n_in; (void)out_size; (void)ws_size;
  const float* X     = (const float*)d_in[0];
  const float* mask  = (const float*)d_in[1];
  const float* Wq    = (const float*)d_in[2];
  const float* bq    = (const float*)d_in[3];
  const float* Wv    = (const float*)d_in[4];
  const float* bv    = (const float*)d_in[5];
  const float* Wo    = (const float*)d_in[6];
  const float* bo    = (const float*)d_in[7];
  const float* Wland = (const float*)d_in[8];
  const float* lng   = (const float*)d_in[9];
  const float* lnb   = (const float*)d_in[10];
  const float* Wconv = (const float*)d_in[11];
  float* Y = (float*)d_out;

  float* ws = (float*)d_ws;
  size_t off = 0;
  float* QS   = ws + off; off += (size_t)BHn * Nn * HDn;   // 25.2M
  float* VB   = ws + off; off += (size_t)BHn * Nn * HDn;   // 25.2M
  float* QL   = ws + off; off += (size_t)BHn * Mn * HDn;
  float* QLN  = ws + off; off += (size_t)BHn * Mn;
  float* K2   = ws + off; off += (size_t)BHn * Mn * Mn;
  float* INVm = ws + off; off += (size_t)BHn * Mn * Mn;
  float* Tg   = ws + off; off += (size_t)BHn * Mn * HDn;
  float* U    = ws + off; off += (size_t)BHn * Mn * HDn;
  float* PRE  = ws + off; off += (size_t)Bn * Nn * Dn;     // 25.2M
  float* SC   = ws + off; off += 16;

  const size_t sh_proj    = (128 * 33 + 2 * 32 * 64) * sizeof(float);
  const size_t sh_land    = (128 * 33 + 32 * 64 + 128 * 64) * sizeof(float);
  const size_t sh_k2      = (128 * 65 + 128 * 128 + 128) * sizeof(float);
  const size_t sh_pnorm   = (128 * 129 + 128 * 128) * sizeof(float);
  const size_t sh_newton  = (3 * 128 * 129) * sizeof(float);
  const size_t sh_ktv     = (3 * 128 * 65 + 128 * 130 + 256) * sizeof(float);
  const size_t sh_invt    = (128 * 129 + 128 * 65) * sizeof(float);
  const size_t sh_out     = (3 * 128 * 65 + 130 * 65 + 128 * 130 + 256) * sizeof(float);
  const size_t sh_outproj = (128 * 33 + 32 * 64) * sizeof(float);

  proj_qv<<<dim3((Bn * Nn) / 128, Hn), 256, sh_proj, stream>>>(
      X, mask, Wq, bq, Wv, bv, QS, VB, SC);
  landmarks_kernel<<<BHn, 256, sh_land, stream>>>(QS, Wland, lng, lnb, QL, QLN);
  k2_kernel<<<BHn, 256, sh_k2, stream>>>(QL, QLN, K2, SC);
  alpha0_kernel<<<1, 1, 0, stream>>>(SC);
  pnorm_kernel<<<BHn, 256, sh_pnorm, stream>>>(K2, SC);
  alpha_final_kernel<<<1, 1, 0, stream>>>(SC);
  newton_kernel<<<BHn, 256, sh_newton, stream>>>(K2, SC, INVm);
  ktv_kernel<<<BHn, 256, sh_ktv, stream>>>(QS, VB, QL, QLN, Tg);
  invt_kernel<<<BHn, 256, sh_invt, stream>>>(INVm, Tg, U);
  out_kernel<<<BHn, 256, sh_out, stream>>>(QS, VB, QL, QLN, U, Wconv, PRE);
  outproj_kernel<<<dim3((Bn * Nn) / 128, Hn), 256, sh_outproj, stream>>>(
      PRE, Wo, bo, Y);
}